// ST_GP_CPAB_47510928228653
// MI455X (gfx1250) — compile-verified
//
#include <hip/hip_runtime.h>
#include <math.h>

// ---------------------------------------------------------------------------
// CPAB 1D spatial transformer for MI455X (gfx1250, wave32)
//   Stage 1: A[b,k] = sum_d theta[b,d] * basis[k,d]  via V_WMMA_F32_16X16X4_F32
//   Stage 2: 64-step Euler ODE (LDS-cached coeffs) + float4 gather-interp
// Problem sizes are fixed by the harness (setup_inputs).
// ---------------------------------------------------------------------------

#define BATCH   256
#define LIN     2048
#define CH      32
#define OUTLEN  2048
#define NCELL   32
#define DDIM    33      // theta / basis inner dim
#define KPAD    36      // DDIM padded to multiple of 4 for K=4 WMMA
#define NSTEPS  64

typedef __attribute__((ext_vector_type(2))) float v2f;
typedef __attribute__((ext_vector_type(8))) float v8f;

// ---------------------------------------------------------------------------
// Stage 1: per-cell affine coefficients via f32 WMMA.
// One wave32 per 16x16 output tile of Aout[256,64]. 16 M-tiles x 4 N-tiles.
//   A operand (theta tile, 16x4 f32):  lane L -> M = L&15,
//       VGPR0 = K = (L<16 ? 0 : 2), VGPR1 = K = (L<16 ? 1 : 3)
//   B operand (basis^T tile, 4x16):    lane L -> N = L&15, same K striping
//   C/D (16x16 f32, 8 VGPRs): VGPR r, lane L -> M = r + (L>=16)*8, N = L&15
// ---------------------------------------------------------------------------
__global__ __launch_bounds__(128)
void cpab_coeff_wmma(const float* __restrict__ theta,   // [256,33]
                     const float* __restrict__ basis,   // [64,33]
                     float* __restrict__ Aout)          // [256,64]
{
    const int lane = threadIdx.x & 31;
    const int wave = (blockIdx.x * (blockDim.x >> 5)) + (threadIdx.x >> 5);
    const int mtile = wave >> 2;        // 0..15  (batch tiles)
    const int ntile = wave & 3;         // 0..3   (2*nC tiles)

    const int l15   = lane & 15;
    const int kHalf = (lane >> 4) * 2;  // 0 or 2

    const int mGlob = mtile * 16 + l15; // batch row this lane feeds for A
    const int nGlob = ntile * 16 + l15; // output column this lane feeds for B

    v8f acc = {};

    #pragma unroll
    for (int kk = 0; kk < KPAD; kk += 4) {
        const int k0 = kk + kHalf;
        v2f a, b;
        a.x = (k0     < DDIM) ? theta[mGlob * DDIM + k0    ] : 0.0f;
        a.y = (k0 + 1 < DDIM) ? theta[mGlob * DDIM + k0 + 1] : 0.0f;
        b.x = (k0     < DDIM) ? basis[nGlob * DDIM + k0    ] : 0.0f;
        b.y = (k0 + 1 < DDIM) ? basis[nGlob * DDIM + k0 + 1] : 0.0f;
        // 8 args: (neg_a, A, neg_b, B, c_mod, C, reuse_a, reuse_b)
        acc = __builtin_amdgcn_wmma_f32_16x16x4_f32(
                  false, a, false, b, (short)0, acc, false, false);
    }

    const int mBase = mtile * 16 + (lane >> 4) * 8;
    #pragma unroll
    for (int r = 0; r < 8; ++r)
        Aout[(mBase + r) * 64 + ntile * 16 + l15] = acc[r];
}

// ---------------------------------------------------------------------------
// Stage 2: ODE integration + gather/interp.
// One block (256 threads, 8 waves) = one batch row x 256 output positions.
//   Phase A: load 32 (a,b) cell pairs into LDS as float2 (ds_load_b64,
//            conflict-free: distinct cells -> distinct bank pairs).
//   Phase B: each thread integrates one phi (64 fused-FMA Euler steps).
//   Phase C: 8 threads per output position, float4 (B128) loads of the two
//            neighboring channel rows, FMA blend, float4 coalesced store.
// ---------------------------------------------------------------------------
__global__ __launch_bounds__(256)
void cpab_warp_interp(const float* __restrict__ x,      // [256,2048,32]
                      const float* __restrict__ Acoef,  // [256,64] (a,b pairs)
                      float* __restrict__ out)          // [256,2048,32]
{
    __shared__ float2 sAB[NCELL];    // per-cell (a, b)
    __shared__ float  sPhi[256];

    const int b   = blockIdx.x >> 3;      // batch row (8 j-tiles per batch)
    const int jt  = blockIdx.x & 7;       // which 256-wide tile of out_len
    const int tid = threadIdx.x;

    if (tid < NCELL) {
        const float2* ap = (const float2*)(Acoef + b * 64);
        sAB[tid] = ap[tid];
    }
    __syncthreads();

    // ---- Euler integration of dphi/dt = a_c*phi + b_c ----
    const int j = jt * 256 + tid;
    float phi = (float)j * (1.0f / (float)(OUTLEN - 1));
    const float dt = 1.0f / (float)NSTEPS;

    #pragma unroll 8
    for (int s = 0; s < NSTEPS; ++s) {
        int c = (int)floorf(phi * (float)NCELL);
        c = min(max(c, 0), NCELL - 1);
        float2 ab = sAB[c];
        phi = fmaf(fmaf(ab.x, phi, ab.y), dt, phi);   // phi += (a*phi+b)*dt
    }
    sPhi[tid] = phi;
    __syncthreads();

    // ---- Linear resample: 8 threads x float4 cover the 32 channels ----
    const int cg  = tid & 7;    // channel group (float4 index)
    const int jl0 = tid >> 3;   // 0..31

    const size_t xbase = (size_t)b * LIN * CH;
    const size_t obase = (size_t)b * OUTLEN * CH;

    #pragma unroll
    for (int pass = 0; pass < 8; ++pass) {
        const int jl = pass * 32 + jl0;              // 0..255 within tile
        float p = fminf(fmaxf(sPhi[jl], 0.0f), 1.0f) * (float)(LIN - 1);
        int i0 = (int)floorf(p);
        i0 = min(max(i0, 0), LIN - 2);
        const float w = p - (float)i0;

        const float4 g0 = ((const float4*)(x + xbase + (size_t)i0 * CH))[cg];
        const float4 g1 = ((const float4*)(x + xbase + (size_t)(i0 + 1) * CH))[cg];

        float4 r;
        r.x = fmaf(g1.x - g0.x, w, g0.x);
        r.y = fmaf(g1.y - g0.y, w, g0.y);
        r.z = fmaf(g1.z - g0.z, w, g0.z);
        r.w = fmaf(g1.w - g0.w, w, g0.w);

        const int jg = jt * 256 + jl;
        ((float4*)(out + obase + (size_t)jg * CH))[cg] = r;
    }
}

// ---------------------------------------------------------------------------
extern "C" void kernel_launch(void* const* d_in, const int* in_sizes, int n_in,
                              void* d_out, int out_size, void* d_ws, size_t ws_size,
                              hipStream_t stream)
{
    (void)in_sizes; (void)n_in; (void)out_size; (void)ws_size;

    const float* x     = (const float*)d_in[0];   // [256,2048,32]
    const float* theta = (const float*)d_in[1];   // [256,33]
    const float* basis = (const float*)d_in[2];   // [64,33]
    float*       out   = (float*)d_out;           // [256,2048,32]
    float*       Aws   = (float*)d_ws;            // [256,64] scratch

    // Stage 1: 64 tiles, one wave32 each -> 16 blocks x 128 threads (4 waves)
    cpab_coeff_wmma<<<16, 128, 0, stream>>>(theta, basis, Aws);

    // Stage 2: 256 batches x 8 tiles of 256 positions
    cpab_warp_interp<<<BATCH * (OUTLEN / 256), 256, 0, stream>>>(x, Aws, out);
}